// Node_OP_21114059227224
// MI455X (gfx1250) — compile-verified
//
#include <hip/hip_runtime.h>

#define N_NODES 50000
#define N_EDGES 800000
#define EMB     128

typedef __attribute__((ext_vector_type(2))) float v2f;
typedef __attribute__((ext_vector_type(8))) float v8f;

// ---------------------------------------------------------------------------
// Zero the workspace (agg_sum [N,128] + cnt [N]) -- harness poisons ws once
// and never re-zeros, so we must do it every call.
// ---------------------------------------------------------------------------
__global__ __launch_bounds__(256)
void zero_ws(float* __restrict__ p, int n) {
    int i = blockIdx.x * blockDim.x + threadIdx.x;
    int stride = gridDim.x * blockDim.x;
    for (; i < n; i += stride) p[i] = 0.0f;
}

// ---------------------------------------------------------------------------
// One wave32 per edge. Lane l handles columns [4l, 4l+4).
// msg = relu(x[src] + edge_attr[e]*W_edge + b_edge); atomic-add into agg[dst].
// x and agg both fit in the 192MB L2, so the atomics stay on-chip.
// ---------------------------------------------------------------------------
__global__ __launch_bounds__(256)
void scatter_edges(const float* __restrict__ x,
                   const float* __restrict__ edge_attr,
                   const float* __restrict__ W_edge,
                   const float* __restrict__ b_edge,
                   const int*   __restrict__ edge_index,
                   float* __restrict__ agg,
                   float* __restrict__ cnt) {
    int tid  = blockIdx.x * blockDim.x + threadIdx.x;
    int e    = tid >> 5;
    int lane = tid & 31;
    if (e >= N_EDGES) return;

    int   src = edge_index[e];
    int   dst = edge_index[N_EDGES + e];
    float ea  = edge_attr[e];

    int c = lane * 4;
    float4 xv = *(const float4*)(x + (size_t)src * EMB + c);
    float4 wv = *(const float4*)(W_edge + c);
    float4 bv = *(const float4*)(b_edge + c);

    float m0 = fmaxf(fmaf(ea, wv.x, bv.x) + xv.x, 0.0f);
    float m1 = fmaxf(fmaf(ea, wv.y, bv.y) + xv.y, 0.0f);
    float m2 = fmaxf(fmaf(ea, wv.z, bv.z) + xv.z, 0.0f);
    float m3 = fmaxf(fmaf(ea, wv.w, bv.w) + xv.w, 0.0f);

    float* ap = agg + (size_t)dst * EMB + c;
    atomicAdd(ap + 0, m0);
    atomicAdd(ap + 1, m1);
    atomicAdd(ap + 2, m2);
    atomicAdd(ap + 3, m3);
    if (lane == 0) atomicAdd(cnt + dst, 1.0f);
}

// ---------------------------------------------------------------------------
// Fused: h = (1+eps)*x + agg/max(cnt,1);  out = relu(h@W1+b1)@W2+b2
// 4 waves/block, each wave owns 16 rows. FP32 WMMA 16x16x4 keeps exact
// fp32 precision (compute is negligible; no reason to downcast).
// ---------------------------------------------------------------------------
#define LDS_STRIDE 130   // pad 128->130: conflict-free A-fragment reads, 8B aligned

__global__ __launch_bounds__(128)
void gin_mlp(const float* __restrict__ x,
             const float* __restrict__ agg,
             const float* __restrict__ cnt,
             const float* __restrict__ W1,
             const float* __restrict__ b1,
             const float* __restrict__ W2,
             const float* __restrict__ b2,
             const float* __restrict__ eps_p,
             const int*   __restrict__ act_p,
             float* __restrict__ out) {
    __shared__ float lds[4][16 * LDS_STRIDE];   // 33,280 bytes

    int w    = threadIdx.x >> 5;
    int lane = threadIdx.x & 31;
    int row0 = blockIdx.x * 64 + w * 16;

    float eps1    = 1.0f + eps_p[0];
    int   add_act = act_p[0];

    float* hb = lds[w];   // wave-private tile: h, then overwritten with t

    // ---- Phase A: build h tile in LDS (uniform branches only) ----
    {
        int c = lane * 4;
        for (int r = 0; r < 16; ++r) {
            int gr = row0 + r;
            float h0 = 0.f, h1 = 0.f, h2 = 0.f, h3 = 0.f;
            if (gr < N_NODES) {                 // uniform within wave
                float4 xv = *(const float4*)(x   + (size_t)gr * EMB + c);
                float4 av = *(const float4*)(agg + (size_t)gr * EMB + c);
                float inv = 1.0f / fmaxf(cnt[gr], 1.0f);
                h0 = fmaf(eps1, xv.x, av.x * inv);
                h1 = fmaf(eps1, xv.y, av.y * inv);
                h2 = fmaf(eps1, xv.z, av.z * inv);
                h3 = fmaf(eps1, xv.w, av.w * inv);
            }
            *(float2*)(hb + r * LDS_STRIDE + c)     = make_float2(h0, h1);
            *(float2*)(hb + r * LDS_STRIDE + c + 2) = make_float2(h2, h3);
        }
    }
    // wave-private LDS tile: DS ops are in-order within a wave, no barrier.

    int m16 = lane & 15;          // A row / B col / D col index
    int kb  = (lane >> 4) << 1;   // K sub-pair base for A/B fragments

    // ---- GEMM1: t = relu(h @ W1 + b1), all 8 N-tiles live in registers ----
    v8f acc[8] = {};
    for (int kt = 0; kt < 32; ++kt) {
        v2f a = *(const v2f*)(hb + m16 * LDS_STRIDE + kt * 4 + kb);
        const float* wp = W1 + (size_t)(kt * 4 + kb) * EMB;
#pragma unroll
        for (int nt = 0; nt < 8; ++nt) {
            v2f b;
            b.x = wp[nt * 16 + m16];
            b.y = wp[EMB + nt * 16 + m16];
            acc[nt] = __builtin_amdgcn_wmma_f32_16x16x4_f32(
                false, a, false, b, (short)0, acc[nt], false, false);
        }
    }
#pragma unroll
    for (int nt = 0; nt < 8; ++nt) {
        float bias = b1[nt * 16 + m16];
#pragma unroll
        for (int v = 0; v < 8; ++v) {
            int m = ((lane >> 4) << 3) + v;     // C/D row layout
            hb[m * LDS_STRIDE + nt * 16 + m16] = fmaxf(acc[nt][v] + bias, 0.0f);
        }
    }

    // ---- GEMM2: out = t @ W2 + b2 ----
    v8f acc2[8] = {};
    for (int kt = 0; kt < 32; ++kt) {
        v2f a = *(const v2f*)(hb + m16 * LDS_STRIDE + kt * 4 + kb);
        const float* wp = W2 + (size_t)(kt * 4 + kb) * EMB;
#pragma unroll
        for (int nt = 0; nt < 8; ++nt) {
            v2f b;
            b.x = wp[nt * 16 + m16];
            b.y = wp[EMB + nt * 16 + m16];
            acc2[nt] = __builtin_amdgcn_wmma_f32_16x16x4_f32(
                false, a, false, b, (short)0, acc2[nt], false, false);
        }
    }
#pragma unroll
    for (int nt = 0; nt < 8; ++nt) {
        float bias = b2[nt * 16 + m16];
#pragma unroll
        for (int v = 0; v < 8; ++v) {
            int m  = ((lane >> 4) << 3) + v;
            int gr = row0 + m;
            float val = acc2[nt][v] + bias;
            if (add_act) val = fmaxf(val, 0.0f);
            if (gr < N_NODES)                   // divergent only AFTER all WMMAs
                out[(size_t)gr * EMB + nt * 16 + m16] = val;
        }
    }
}

// ---------------------------------------------------------------------------
extern "C" void kernel_launch(void* const* d_in, const int* in_sizes, int n_in,
                              void* d_out, int out_size, void* d_ws, size_t ws_size,
                              hipStream_t stream) {
    const float* x         = (const float*)d_in[0];
    const float* edge_attr = (const float*)d_in[1];
    const float* W_edge    = (const float*)d_in[2];
    const float* b_edge    = (const float*)d_in[3];
    const float* W1        = (const float*)d_in[4];
    const float* b1        = (const float*)d_in[5];
    const float* W2        = (const float*)d_in[6];
    const float* b2        = (const float*)d_in[7];
    const float* eps       = (const float*)d_in[8];
    const int*   edge_idx  = (const int*)d_in[9];
    const int*   add_act   = (const int*)d_in[10];
    float* out = (float*)d_out;

    float* agg = (float*)d_ws;                       // [N_NODES, EMB]
    float* cnt = agg + (size_t)N_NODES * EMB;        // [N_NODES]

    int nz = N_NODES * EMB + N_NODES;
    zero_ws<<<1024, 256, 0, stream>>>(agg, nz);

    // 800000 edges * 32 lanes / 256 threads = 100000 blocks
    scatter_edges<<<(N_EDGES * 32) / 256, 256, 0, stream>>>(
        x, edge_attr, W_edge, b_edge, edge_idx, agg, cnt);

    gin_mlp<<<(N_NODES + 63) / 64, 128, 0, stream>>>(
        x, agg, cnt, W1, b1, W2, b2, eps, add_act, out);
}